// Pair_attention_36283883717042
// MI455X (gfx1250) — compile-verified
//
#include <hip/hip_runtime.h>
#include <hip/hip_bf16.h>

typedef __attribute__((ext_vector_type(2))) float v2f;
typedef __attribute__((ext_vector_type(4))) float v4f;
typedef __attribute__((ext_vector_type(8))) float v8f;

#define EMBED 1024
#define HEADS 16
#define HEAD_DIM 64
#define NSEQ 4096
#define ROWPAD 4100          // 4096 + 4 floats: LDS bank-conflict padding

static __device__ __forceinline__ v8f wmma_f32x4(v2f a, v2f b, v8f c) {
    return __builtin_amdgcn_wmma_f32_16x16x4_f32(
        false, a, false, b, (short)0, c, false, false);
}

// ---------------------------------------------------------------------------
// Kernel 1: q/k/v projections.  One wave per (n-tile, head, which).
// ---------------------------------------------------------------------------
__global__ __launch_bounds__(32)
void qkv_kernel(const float* __restrict__ pairs,
                const float* __restrict__ Wq,
                const float* __restrict__ Wk,
                const float* __restrict__ Wv,
                float* __restrict__ q,
                float* __restrict__ k,
                float* __restrict__ v) {
    const int nt = blockIdx.x;
    const int h  = blockIdx.y;
    const int which = blockIdx.z;
    const float* W = (which == 0) ? Wq : (which == 1) ? Wk : Wv;
    float* dst     = (which == 0) ? q  : (which == 1) ? k  : v;

    const int lm = threadIdx.x & 15;
    const int lh = threadIdx.x >> 4;

    const float* arow = pairs + (size_t)(nt * 16 + lm) * EMBED + h * HEAD_DIM + lh * 2;

    v8f c0 = {}, c1 = {}, c2 = {}, c3 = {};
#pragma unroll
    for (int t = 0; t < 16; ++t) {
        const int d = 4 * t;
        v2f a  = *(const v2f*)(arow + d);
        v2f b0 = *(const v2f*)(W + (size_t)( 0 + lm) * HEAD_DIM + d + lh * 2);
        v2f b1 = *(const v2f*)(W + (size_t)(16 + lm) * HEAD_DIM + d + lh * 2);
        v2f b2 = *(const v2f*)(W + (size_t)(32 + lm) * HEAD_DIM + d + lh * 2);
        v2f b3 = *(const v2f*)(W + (size_t)(48 + lm) * HEAD_DIM + d + lh * 2);
        c0 = wmma_f32x4(a, b0, c0);
        c1 = wmma_f32x4(a, b1, c1);
        c2 = wmma_f32x4(a, b2, c2);
        c3 = wmma_f32x4(a, b3, c3);
    }
#pragma unroll
    for (int r = 0; r < 8; ++r) {
        float* o = dst + (size_t)(nt * 16 + r + lh * 8) * EMBED + h * HEAD_DIM + lm;
        o[0]  = c0[r];
        o[16] = c1[r];
        o[32] = c2[r];
        o[48] = c3[r];
    }
}

// ---------------------------------------------------------------------------
// Kernel 2 (fused): energy -> softmax -> ctx for one (q-tile, head).
// 512 threads = 16 waves.  16x4096 energy strip lives in LDS (320KB WGP LDS).
//   Phase 1: WMMA energy strip -> LDS only
//   Phase 2: per-row wave: b128 NT-store energy to HBM while computing max,
//            then exp/sum, then normalize -> LDS + b128 NT-store attention
//   Phase 3: ctx strip = att_strip @ V_head (V is L2-resident)
// ---------------------------------------------------------------------------
__global__ __launch_bounds__(512)
void fused_attention_kernel(const float* __restrict__ q,
                            const float* __restrict__ k,
                            const float* __restrict__ v,
                            float* __restrict__ energy,
                            float* __restrict__ att,
                            float* __restrict__ ctx) {
    __shared__ __align__(16) float strip[16 * ROWPAD];   // 262400 B
    __shared__ float scratch[16 * 256];                  //  16384 B

    const int qt = blockIdx.x;
    const int h  = blockIdx.y;
    const int tid  = threadIdx.x;
    const int w    = tid >> 5;       // wave id 0..15
    const int lane = tid & 31;
    const int lm   = lane & 15;
    const int lh   = lane >> 4;

    // ---------------- Phase 1: energy strip (LDS only) ----------------
    {
        const float* qrow = q + (size_t)(qt * 16 + lm) * EMBED + h * HEAD_DIM + lh * 2;
        v2f a[16];
#pragma unroll
        for (int t = 0; t < 16; ++t) a[t] = *(const v2f*)(qrow + 4 * t);

        for (int i = 0; i < 16; ++i) {           // 16 k-tiles per wave
            const int kt = w * 16 + i;
            const float* krow = k + (size_t)(kt * 16 + lm) * EMBED + h * HEAD_DIM + lh * 2;
            v8f c = {};
#pragma unroll
            for (int t = 0; t < 16; ++t) {
                v2f b = *(const v2f*)(krow + 4 * t);
                c = wmma_f32x4(a[t], b, c);
            }
#pragma unroll
            for (int r = 0; r < 8; ++r)
                strip[(r + lh * 8) * ROWPAD + kt * 16 + lm] = c[r];
        }
    }
    __syncthreads();

    // ---------------- Phase 2: softmax (one wave per row) ----------------
    {
        const float scale = 0.03125f;            // 1/sqrt(1024)
        float* row  = strip + w * ROWPAD;
        v4f*   row4 = (v4f*)row;
        v4f*   erow4 = (v4f*)(energy + ((size_t)h * NSEQ + qt * 16 + w) * NSEQ);
        v4f*   arow4 = (v4f*)(att    + ((size_t)h * NSEQ + qt * 16 + w) * NSEQ);

        // pass 1: stream energy row to HBM (b128 NT) while computing max
        float m = -3.402823466e+38f;
#pragma unroll 4
        for (int i = 0; i < 32; ++i) {
            v4f x = row4[lane + i * 32];
            __builtin_nontemporal_store(x, erow4 + lane + i * 32);
            m = fmaxf(m, fmaxf(fmaxf(x.x, x.y), fmaxf(x.z, x.w)));
        }
#pragma unroll
        for (int off = 16; off >= 1; off >>= 1)
            m = fmaxf(m, __shfl_xor(m, off, 32));
        m *= scale;

        // pass 2: exp + sum (overwrite LDS row with exponentials)
        float sum = 0.0f;
#pragma unroll 4
        for (int i = 0; i < 32; ++i) {
            v4f x = row4[lane + i * 32];
            x.x = __expf(x.x * scale - m);
            x.y = __expf(x.y * scale - m);
            x.z = __expf(x.z * scale - m);
            x.w = __expf(x.w * scale - m);
            sum += (x.x + x.y) + (x.z + x.w);
            row4[lane + i * 32] = x;
        }
#pragma unroll
        for (int off = 16; off >= 1; off >>= 1)
            sum += __shfl_xor(sum, off, 32);
        const float inv = 1.0f / sum;

        // pass 3: normalize -> LDS (for phase 3) + b128 NT store to HBM
#pragma unroll 4
        for (int i = 0; i < 32; ++i) {
            v4f x = row4[lane + i * 32];
            x *= inv;
            row4[lane + i * 32] = x;
            __builtin_nontemporal_store(x, arow4 + lane + i * 32);
        }
    }
    __syncthreads();

    // ---------------- Phase 3: ctx strip = att @ V_head ----------------
    {
        const int j  = w & 3;                    // d-tile (16 cols of HEAD_DIM)
        const int kq = w >> 2;                   // K-quarter
        const int kbase = kq * 1024;

        v8f c = {};
        for (int kk = kbase; kk < kbase + 1024; kk += 4) {
            v2f a = *(const v2f*)&strip[lm * ROWPAD + kk + lh * 2];
            const int kr = kk + lh * 2;
            const float* vb = v + (size_t)kr * EMBED + h * HEAD_DIM + j * 16 + lm;
            v2f b = { vb[0], vb[EMBED] };
            c = wmma_f32x4(a, b, c);
        }
#pragma unroll
        for (int r = 0; r < 8; ++r)
            scratch[w * 256 + r * 32 + lane] = c[r];
    }
    __syncthreads();

    // combine the 4 K-quarter partials; waves 0..3 each own one d-tile
    if (w < 4) {
#pragma unroll
        for (int r = 0; r < 8; ++r) {
            float s = scratch[(0 * 4 + w) * 256 + r * 32 + lane]
                    + scratch[(1 * 4 + w) * 256 + r * 32 + lane]
                    + scratch[(2 * 4 + w) * 256 + r * 32 + lane]
                    + scratch[(3 * 4 + w) * 256 + r * 32 + lane];
            ctx[(size_t)(qt * 16 + r + lh * 8) * EMBED + h * HEAD_DIM + w * 16 + lm] = s;
        }
    }
}

// ---------------------------------------------------------------------------
// Kernel 3: out = ctx @ Wo^T + bo.  One wave per 16x64 output strip
// (A-fragment reused across 4 column tiles).
// ---------------------------------------------------------------------------
__global__ __launch_bounds__(32)
void outproj_kernel(const float* __restrict__ ctx,
                    const float* __restrict__ Wo,
                    const float* __restrict__ bo,
                    float* __restrict__ out) {
    const int qt = blockIdx.x;
    const int eg = blockIdx.y;                   // group of 4 e-tiles (64 cols)
    const int lm = threadIdx.x & 15;
    const int lh = threadIdx.x >> 4;

    const float* arow = ctx + (size_t)(qt * 16 + lm) * EMBED + lh * 2;
    const float* b0r  = Wo + (size_t)(eg * 64 +  0 + lm) * EMBED + lh * 2;
    const float* b1r  = Wo + (size_t)(eg * 64 + 16 + lm) * EMBED + lh * 2;
    const float* b2r  = Wo + (size_t)(eg * 64 + 32 + lm) * EMBED + lh * 2;
    const float* b3r  = Wo + (size_t)(eg * 64 + 48 + lm) * EMBED + lh * 2;

    v8f c0 = {}, c1 = {}, c2 = {}, c3 = {};
#pragma unroll 4
    for (int cc = 0; cc < EMBED; cc += 4) {
        v2f a  = *(const v2f*)(arow + cc);
        v2f b0 = *(const v2f*)(b0r + cc);
        v2f b1 = *(const v2f*)(b1r + cc);
        v2f b2 = *(const v2f*)(b2r + cc);
        v2f b3 = *(const v2f*)(b3r + cc);
        c0 = wmma_f32x4(a, b0, c0);
        c1 = wmma_f32x4(a, b1, c1);
        c2 = wmma_f32x4(a, b2, c2);
        c3 = wmma_f32x4(a, b3, c3);
    }
#pragma unroll
    for (int r = 0; r < 8; ++r) {
        float* o = out + (size_t)(qt * 16 + r + lh * 8) * EMBED + eg * 64 + lm;
        o[0]  = c0[r] + bo[eg * 64 +  0 + lm];
        o[16] = c1[r] + bo[eg * 64 + 16 + lm];
        o[32] = c2[r] + bo[eg * 64 + 32 + lm];
        o[48] = c3[r] + bo[eg * 64 + 48 + lm];
    }
}

// ---------------------------------------------------------------------------
extern "C" void kernel_launch(void* const* d_in, const int* in_sizes, int n_in,
                              void* d_out, int out_size, void* d_ws, size_t ws_size,
                              hipStream_t stream) {
    const float* pairs = (const float*)d_in[0];
    const float* Wq    = (const float*)d_in[1];
    const float* Wk    = (const float*)d_in[2];
    const float* Wv    = (const float*)d_in[3];
    const float* Wo    = (const float*)d_in[4];
    const float* bo    = (const float*)d_in[5];

    // Output layout: out [N,EMBED] | energy [H,N,N] | attention [H,N,N]
    float* out    = (float*)d_out;
    float* energy = out + (size_t)NSEQ * EMBED;
    float* att    = energy + (size_t)HEADS * NSEQ * NSEQ;

    // Workspace: q | k | v | ctx  (16 MB each)
    float* q   = (float*)d_ws;
    float* k   = q + (size_t)NSEQ * EMBED;
    float* v   = k + (size_t)NSEQ * EMBED;
    float* ctx = v + (size_t)NSEQ * EMBED;

    qkv_kernel<<<dim3(NSEQ / 16, HEADS, 3), dim3(32), 0, stream>>>(pairs, Wq, Wk, Wv, q, k, v);
    fused_attention_kernel<<<dim3(NSEQ / 16, HEADS), dim3(512), 0, stream>>>(
        q, k, v, energy, att, ctx);
    outproj_kernel<<<dim3(NSEQ / 16, EMBED / 64), dim3(32), 0, stream>>>(ctx, Wo, bo, out);
}